// RoIAlign_73581379715142
// MI455X (gfx1250) — compile-verified
//
#include <hip/hip_runtime.h>
#include <stdint.h>
#include <math.h>

// RoIAlign (N=512, C=256, H=W=32, S=7) for MI455X / gfx1250.
//
// Memory-bound gather (AI ~0.5 flop/byte): floor = 512MB read + 25.7MB write
// at 23.3 TB/s ~= 23 us. Strategy: per block, TDM-DMA a 64KB block of 16
// channel images (this RoI's data) into LDS, then do all bilinear gathers +
// 2x2 max-pool from LDS. WMMA does not pay here (FLOPs are negligible).

#define N_MAX   512
#define C_TOT   256
#define HH      32
#define WW      32
#define S7      7
#define CPB     16          // channels staged per block
#define GRID_G  14          // 2*S sample grid

typedef unsigned int u32;
typedef u32 __attribute__((ext_vector_type(4))) u32x4;
typedef int  __attribute__((ext_vector_type(8))) i32x8;
typedef int  __attribute__((ext_vector_type(4))) i32x4;

__global__ __launch_bounds__(256) void roialign_tdm_kernel(
    const float* __restrict__ feature,   // (N, C, H, W) fp32
    const float* __restrict__ bbox,      // (N, 4) fp32
    float* __restrict__ out)             // (N, C, 7, 7) fp32
{
    __shared__ float smem[CPB * HH * WW];   // 64 KB: 16 channels x 32x32

    const int cb = blockIdx.x;            // channel block 0..15
    const int n  = blockIdx.y;            // RoI index
    const int c0 = cb * CPB;

    // ---------------- Stage feature channels into LDS ----------------
#if __has_builtin(__builtin_amdgcn_tensor_load_to_lds)
    if (threadIdx.x < 32) {   // one wave issues the DMA (TDM ignores EXEC)
        const uint64_t gaddr =
            (uint64_t)(const void*)(feature + ((size_t)n * C_TOT + c0) * (HH * WW));
        // Flat pointer into LDS aperture: low 32 bits == LDS byte offset.
        const u32 lds_base = (u32)(uintptr_t)(void*)smem;

        // D# group 0 (ISA 08 Async/Tensor, sec 8.3)
        u32x4 g0;
        g0[0] = 1u;                                   // count=1, user descriptor
        g0[1] = lds_base;                             // lds_addr (bytes)
        g0[2] = (u32)(gaddr & 0xffffffffu);           // global_addr[31:0]
        g0[3] = (u32)((gaddr >> 32) & 0x01ffffffu)    // global_addr[56:32]
              | 0x80000000u;                          // type = 2 ("image")

        // D# group 1 (sec 8.4): 2D tile, 4-byte elements,
        //   tensor_dim0 = 1024, tensor_dim1 = 16 (== tile dims, no OOB)
        //   tile_dim0 = 1024, tile_dim1 = 16, tensor_dim0_stride = 1024
        i32x8 g1;
        g1[0] = (int)(2u << 16);                      // data_size=2 (4B); mask/pad/iter=0
        g1[1] = (int)((u32)(HH * WW) << 16);          // tensor_dim0[15:0]
        g1[2] = (int)((u32)CPB << 16);                // tensor_dim0[31:16]=0 | tensor_dim1[15:0]
        g1[3] = (int)((u32)(HH * WW) << 16);          // tensor_dim1[31:16]=0 | tile_dim0
        g1[4] = (int)CPB;                             // tile_dim1 | tile_dim2=0
        g1[5] = (int)(HH * WW);                       // tensor_dim0_stride[31:0]
        g1[6] = 0;                                    // stride0[47:32] | stride1[15:0]
        g1[7] = 0;                                    // stride1[47:16]

        const i32x4 gz4 = {0, 0, 0, 0};               // groups 2/3 unused (2D tensor)
        const i32x8 gz8 = {0, 0, 0, 0, 0, 0, 0, 0};   // extra slot (6-arg toolchain)
        __builtin_amdgcn_tensor_load_to_lds(g0, g1, gz4, gz4, gz8, 0);
        __builtin_amdgcn_s_wait_tensorcnt(0);         // wave-local DMA completion
    }
#else
    // Fallback: cooperative 128-bit staging loads.
    {
        const float4* src =
            (const float4*)(feature + ((size_t)n * C_TOT + c0) * (HH * WW));
        float4* dst = (float4*)smem;
        for (int i = threadIdx.x; i < CPB * HH * WW / 4; i += blockDim.x)
            dst[i] = src[i];
    }
#endif

    // bbox / sample-grid parameters (uniform per block)
    const float bx1 = bbox[4 * n + 0];
    const float by1 = bbox[4 * n + 1];
    const float bx2 = bbox[4 * n + 2];
    const float by2 = bbox[4 * n + 3];
    const float xstep = (bx2 - bx1) * (1.0f / (float)GRID_G);
    const float ystep = (by2 - by1) * (1.0f / (float)GRID_G);

    __syncthreads();   // LDS tile visible to all 8 waves

    // ------------- Bilinear sample + 2x2 max-pool from LDS -------------
    // 784 work items (16 ch x 7x7); lane varies fastest over the 7x7 outputs
    // so LDS addresses within a wave differ in spatial offset, not in the
    // bank-aligned 1024-dword channel stride.
    for (int w = threadIdx.x; w < CPB * S7 * S7; w += blockDim.x) {
        const int c  = w / (S7 * S7);
        const int o  = w - c * (S7 * S7);
        const int oy = o / S7;
        const int ox = o - oy * S7;
        const float* ch = smem + c * (HH * WW);

        float best = -INFINITY;
#pragma unroll
        for (int sy = 0; sy < 2; ++sy) {
            const float ys = by1 + (float)(2 * oy + sy) * ystep;
            float y0f = floorf(ys);
            y0f = fminf(fmaxf(y0f, 0.0f), (float)(HH - 1));
            const int   y0 = (int)y0f;
            const int   y1 = min(y0 + 1, HH - 1);
            const float wy = ys - y0f;
#pragma unroll
            for (int sx = 0; sx < 2; ++sx) {
                const float xs = bx1 + (float)(2 * ox + sx) * xstep;
                float x0f = floorf(xs);
                x0f = fminf(fmaxf(x0f, 0.0f), (float)(WW - 1));
                const int   x0 = (int)x0f;
                const int   x1 = min(x0 + 1, WW - 1);
                const float wx = xs - x0f;

                const float v00 = ch[y0 * WW + x0];
                const float v01 = ch[y0 * WW + x1];
                const float v10 = ch[y1 * WW + x0];
                const float v11 = ch[y1 * WW + x1];

                const float top = v00 + wx * (v01 - v00);
                const float bot = v10 + wx * (v11 - v10);
                const float v   = top + wy * (bot - top);
                best = fmaxf(best, v);
            }
        }
        out[(((size_t)n * C_TOT + c0 + c) * S7 + oy) * S7 + ox] = best;
    }
}

extern "C" void kernel_launch(void* const* d_in, const int* in_sizes, int n_in,
                              void* d_out, int out_size, void* d_ws, size_t ws_size,
                              hipStream_t stream) {
    const float* feature = (const float*)d_in[0];
    const float* bbox    = (const float*)d_in[1];
    float*       out     = (float*)d_out;

    const int N = in_sizes[1] / 4;            // 512
    dim3 grid(C_TOT / CPB, N);                // (16, 512) blocks
    roialign_tdm_kernel<<<grid, 256, 0, stream>>>(feature, bbox, out);
    (void)n_in; (void)out_size; (void)d_ws; (void)ws_size;
}